// GAE_38517266710716
// MI455X (gfx1250) — compile-verified
//
#include <hip/hip_runtime.h>
#include <hip/hip_bf16.h>

// ---------------------------------------------------------------------------
// GAE pipeline for MI455X (gfx1250), wave32.
//   N = 7*4096 = 28672 nodes, F = 256, E = 458752 edges.
//   encoder gconv 256->256->128->64 (+BN/ReLU), conv1d codec on 64-wide rows,
//   decoder gconv 64->128->256->256 (+BN/ReLU).
// GEMMs use V_WMMA_F32_16X16X4_F32 with weight panels staged in LDS
// (fragment-order float2 pairs -> one ds_load_b64 per B fragment).
// Edge scatter uses global_atomic_add_f32 via unsafeAtomicAdd.
// ---------------------------------------------------------------------------

#define NN 28672
#define FF 256
#define EE 458752
#define KC 64          // K-chunk staged in LDS per GEMM block

typedef __attribute__((ext_vector_type(2))) float v2f;
typedef __attribute__((ext_vector_type(8))) float v8f;

struct Ptr7 { const float* p[7]; };

// ---- build x (N,256) from the 7 inputs; x row (b*7+j) = xj row b.
// Written straight into d_out[0 .. N*F) == xo.
__global__ void build_x_kernel(float* __restrict__ x, Ptr7 xs) {
  int idx = blockIdx.x * 256 + threadIdx.x;       // exactly NN*FF threads
  int n = idx >> 8;
  int f = idx & 255;
  int b = n / 7, j = n - b * 7;
  x[idx] = xs.p[j][(size_t)b * FF + f];
}

__global__ void zero_kernel(float* __restrict__ p, int n) {
  int i = blockIdx.x * 256 + threadIdx.x;
  if (i < n) p[i] = 0.0f;
}

// ---- edge scatter: agg[dst] += h[src] * ew     (float4 per thread)
__global__ void scatter_kernel(const float* __restrict__ h,
                               const int* __restrict__ src,
                               const int* __restrict__ dst,
                               const float* __restrict__ ew,
                               float* __restrict__ agg,
                               int di, int csh) {
  long long t = (long long)blockIdx.x * 256 + threadIdx.x;
  long long total = (long long)EE << csh;          // E * (di/4)
  if (t >= total) return;
  int e = (int)(t >> csh);
  int c = (int)(t & ((1 << csh) - 1));
  float w = ew[e];
  const float4 v = *((const float4*)(h + (size_t)src[e] * di) + c);
  float* ap = agg + (size_t)dst[e] * di + (c << 2);
  unsafeAtomicAdd(ap + 0, v.x * w);
  unsafeAtomicAdd(ap + 1, v.y * w);
  unsafeAtomicAdd(ap + 2, v.z * w);
  unsafeAtomicAdd(ap + 3, v.w * w);
}

// ---- dual GEMM + bias:  out = A1 @ W1 + A2 @ W2 + bias
//   A1/A2: (NN x di) row-major; W1/W2: (di x dO) row-major.
//   Per wave: 16 rows x 64 cols, K-loop step 4 using V_WMMA_F32_16X16X4_F32.
//   A 16x4 layout (ISA 7.12.2): lanes 0-15 / 16-31 give M, VGPR0/1 = K pairs
//   {0,1} (low lanes) or {2,3} (high lanes). B 4x16 mirrored; C/D per table.
//   Weight panels (KC x 64, both matrices) staged in LDS in fragment order:
//   lB[j*64 + n] = (W[kc+2j][n0+n], W[kc+2j+1][n0+n]) -> one ds_load_b64.
__global__ void gemm_wmma_kernel(const float* __restrict__ A1,
                                 const float* __restrict__ A2,
                                 const float* __restrict__ W1,
                                 const float* __restrict__ W2,
                                 const float* __restrict__ bias,
                                 float* __restrict__ outp,
                                 int di, int dO) {
  __shared__ float2 lB1[(KC / 2) * 64];   // 16 KB
  __shared__ float2 lB2[(KC / 2) * 64];   // 16 KB
  const int tid  = threadIdx.x;
  const int wave = tid >> 5;
  const int lane = tid & 31;
  const int hf   = lane >> 4;       // 0: K pair {0,1}, 1: K pair {2,3}
  const int ln   = lane & 15;
  const int m0 = (blockIdx.x * 8 + wave) * 16;
  const int n0 = blockIdx.y * 64;

  v8f acc[4];
#pragma unroll
  for (int t = 0; t < 4; t++)
#pragma unroll
    for (int v = 0; v < 8; v++) acc[t][v] = 0.0f;

  const float* a1row = A1 + (size_t)(m0 + ln) * di + hf * 2;
  const float* a2row = A2 + (size_t)(m0 + ln) * di + hf * 2;

  for (int kc = 0; kc < di; kc += KC) {
    __syncthreads();
    // cooperative stage of both 64-col weight panels (coalesced in n)
    for (int idx = tid; idx < (KC / 2) * 64; idx += 256) {
      int j = idx >> 6;            // row pair
      int n = idx & 63;            // column within panel
      const float* p1 = W1 + (size_t)(kc + 2 * j) * dO + n0 + n;
      const float* p2 = W2 + (size_t)(kc + 2 * j) * dO + n0 + n;
      lB1[idx] = make_float2(p1[0], p1[dO]);
      lB2[idx] = make_float2(p2[0], p2[dO]);
    }
    __syncthreads();

    for (int kl = 0; kl < KC; kl += 4) {
      int k = kc + kl;
      v2f a1; a1.x = a1row[k]; a1.y = a1row[k + 1];
      v2f a2; a2.x = a2row[k]; a2.y = a2row[k + 1];
      const int j = (kl >> 1) + hf;            // fragment pair index
      const float2* b1p = &lB1[j * 64 + ln];
      const float2* b2p = &lB2[j * 64 + ln];
#pragma unroll
      for (int t = 0; t < 4; t++) {
        float2 f1 = b1p[t * 16];
        float2 f2 = b2p[t * 16];
        v2f b1; b1.x = f1.x; b1.y = f1.y;
        v2f b2; b2.x = f2.x; b2.y = f2.y;
        acc[t] = __builtin_amdgcn_wmma_f32_16x16x4_f32(
            false, a1, false, b1, (short)0, acc[t], false, false);
        acc[t] = __builtin_amdgcn_wmma_f32_16x16x4_f32(
            false, a2, false, b2, (short)0, acc[t], false, false);
      }
    }
  }

#pragma unroll
  for (int t = 0; t < 4; t++) {
    int col = n0 + t * 16 + ln;
    float bv = bias[col];
#pragma unroll
    for (int v = 0; v < 8; v++) {
      int row = m0 + hf * 8 + v;
      outp[(size_t)row * dO + col] = acc[t][v] + bv;
    }
  }
}

// ---- per-column mean/var over NN rows (one block per column)
__global__ void colstats_kernel(const float* __restrict__ h,
                                float* __restrict__ mean,
                                float* __restrict__ var, int dO) {
  int j = blockIdx.x;
  __shared__ float s1[256];
  __shared__ float s2[256];
  float a = 0.0f, b = 0.0f;
  for (int m = threadIdx.x; m < NN; m += 256) {
    float v = h[(size_t)m * dO + j];
    a += v; b += v * v;
  }
  s1[threadIdx.x] = a; s2[threadIdx.x] = b;
  __syncthreads();
  for (int s = 128; s > 0; s >>= 1) {
    if (threadIdx.x < s) {
      s1[threadIdx.x] += s1[threadIdx.x + s];
      s2[threadIdx.x] += s2[threadIdx.x + s];
    }
    __syncthreads();
  }
  if (threadIdx.x == 0) {
    float mu = s1[0] * (1.0f / NN);
    mean[j] = mu;
    var[j] = s2[0] * (1.0f / NN) - mu * mu;
  }
}

// ---- BN + ReLU in place
__global__ void bnrelu_kernel(float* __restrict__ h,
                              const float* __restrict__ g,
                              const float* __restrict__ b,
                              const float* __restrict__ mean,
                              const float* __restrict__ var, int dO) {
  long long idx = (long long)blockIdx.x * 256 + threadIdx.x;
  if (idx >= (long long)NN * dO) return;
  int j = (int)(idx % dO);
  float v = h[idx];
  v = g[j] * (v - mean[j]) * rsqrtf(var[j] + 1e-5f) + b[j];
  h[idx] = v > 0.0f ? v : 0.0f;
}

// ---- fused conv1d codec (1->8->16->1->16->8->1 channels, k=3, pad=1)
// One 64-thread block per node; LDS ping-pong with 1-element halos.
__device__ __forceinline__ void conv_stage(const float* __restrict__ in,
                                           float* __restrict__ out,
                                           int Cin, int Cout,
                                           const float* __restrict__ w,
                                           const float* __restrict__ b,
                                           bool relu, bool flipT, int tid) {
  for (int idx = tid; idx < Cout * 64; idx += 64) {
    int c = idx >> 6, p = idx & 63;
    float acc = b[c];
    for (int ci = 0; ci < Cin; ci++) {
#pragma unroll
      for (int k = 0; k < 3; k++) {
        // deconv: w2[c][ci][k] = w_orig[ci][c][2-k]  (flip + transpose)
        float wv = flipT ? w[(ci * Cout + c) * 3 + (2 - k)]
                         : w[(c * Cin + ci) * 3 + k];
        acc += wv * in[ci * 66 + p + k];     // halo at 0 and 65 is zero
      }
    }
    out[c * 66 + p + 1] = (relu && acc < 0.0f) ? 0.0f : acc;
  }
}

__global__ void conv_pipe_kernel(const float* __restrict__ hin,
                                 float* __restrict__ hout,
                                 const float* we0, const float* be0,
                                 const float* we1, const float* be1,
                                 const float* we2, const float* be2,
                                 const float* wd0, const float* bd0,
                                 const float* wd1, const float* bd1,
                                 const float* wd2, const float* bd2) {
  __shared__ float bufA[16 * 66];
  __shared__ float bufB[16 * 66];
  const int tid = threadIdx.x;
  const int node = blockIdx.x;
  for (int i = tid; i < 16 * 66; i += 64) { bufA[i] = 0.0f; bufB[i] = 0.0f; }
  __syncthreads();
  bufA[1 + tid] = hin[(size_t)node * 64 + tid];
  __syncthreads();
  conv_stage(bufA, bufB, 1,  8,  we0, be0, true,  false, tid); __syncthreads();
  conv_stage(bufB, bufA, 8,  16, we1, be1, true,  false, tid); __syncthreads();
  conv_stage(bufA, bufB, 16, 1,  we2, be2, false, false, tid); __syncthreads();
  conv_stage(bufB, bufA, 1,  16, wd0, bd0, true,  true,  tid); __syncthreads();
  conv_stage(bufA, bufB, 16, 8,  wd1, bd1, true,  true,  tid); __syncthreads();
  conv_stage(bufB, bufA, 8,  1,  wd2, bd2, false, true,  tid); __syncthreads();
  hout[(size_t)node * 64 + tid] = bufA[1 + tid];
}

// ---------------------------------------------------------------------------
extern "C" void kernel_launch(void* const* d_in, const int* in_sizes, int n_in,
                              void* d_out, int out_size, void* d_ws, size_t ws_size,
                              hipStream_t stream) {
  // Inputs 0-6: x1..x7; 7: edge_index (2,E) int32; 8: edge_attr (E,)
  // params flattened in JAX pytree order (dict keys sorted):
  //   dbn (9-12), dec (13-21), deconv (22-27), ebn (28-31), enc (32-40), enconv (41-46)
  Ptr7 p7;
  for (int i = 0; i < 7; i++) p7.p[i] = (const float*)d_in[i];
  const int* ei  = (const int*)d_in[7];
  const int* src = ei;
  const int* dst = ei + EE;
  const float* ew = (const float*)d_in[8];

  const float* dbn0g = (const float*)d_in[9];   // 128
  const float* dbn0b = (const float*)d_in[10];  // 128
  const float* dbn1g = (const float*)d_in[11];  // 256
  const float* dbn1b = (const float*)d_in[12];  // 256
  const float* dec0wr = (const float*)d_in[13]; // 64x128
  const float* dec0ws = (const float*)d_in[14];
  const float* dec0b  = (const float*)d_in[15];
  const float* dec1wr = (const float*)d_in[16]; // 128x256
  const float* dec1ws = (const float*)d_in[17];
  const float* dec1b  = (const float*)d_in[18];
  const float* dec2wr = (const float*)d_in[19]; // 256x256
  const float* dec2ws = (const float*)d_in[20];
  const float* dec2b  = (const float*)d_in[21];
  const float* wd0 = (const float*)d_in[22];    // (1,16,3)
  const float* bd0 = (const float*)d_in[23];    // 16
  const float* wd1 = (const float*)d_in[24];    // (16,8,3)
  const float* bd1 = (const float*)d_in[25];    // 8
  const float* wd2 = (const float*)d_in[26];    // (8,1,3)
  const float* bd2 = (const float*)d_in[27];    // 1
  const float* ebn0g = (const float*)d_in[28];  // 256
  const float* ebn0b = (const float*)d_in[29];
  const float* ebn1g = (const float*)d_in[30];  // 128
  const float* ebn1b = (const float*)d_in[31];
  const float* enc0wr = (const float*)d_in[32]; // 256x256
  const float* enc0ws = (const float*)d_in[33];
  const float* enc0b  = (const float*)d_in[34];
  const float* enc1wr = (const float*)d_in[35]; // 256x128
  const float* enc1ws = (const float*)d_in[36];
  const float* enc1b  = (const float*)d_in[37];
  const float* enc2wr = (const float*)d_in[38]; // 128x64
  const float* enc2ws = (const float*)d_in[39];
  const float* enc2b  = (const float*)d_in[40];
  const float* we0 = (const float*)d_in[41];    // (8,1,3)
  const float* be0 = (const float*)d_in[42];    // 8
  const float* we1 = (const float*)d_in[43];    // (16,8,3)
  const float* be1 = (const float*)d_in[44];    // 16
  const float* we2 = (const float*)d_in[45];    // (1,16,3)
  const float* be2 = (const float*)d_in[46];    // 1

  float* x   = (float*)d_out;                   // == xo (N,256) row-major
  float* ho  = (float*)d_out + (size_t)NN * FF; // final decoder output
  float* agg = (float*)d_ws;
  float* hA  = agg + (size_t)NN * 256;
  float* hB  = hA  + (size_t)NN * 256;
  float* mean = hB + (size_t)NN * 256;
  float* var  = mean + 256;

  build_x_kernel<<<NN, 256, 0, stream>>>(x, p7);  // NN*256 threads

  auto gconv = [&](const float* hin, float* hout, int di, int dO,
                   const float* wr, const float* wroot, const float* bias) {
    int zn = NN * di;
    zero_kernel<<<(zn + 255) / 256, 256, 0, stream>>>(agg, zn);
    int csh = (di == 256) ? 6 : (di == 128) ? 5 : 4;   // log2(di/4)
    long long tot = (long long)EE << csh;
    scatter_kernel<<<(int)((tot + 255) / 256), 256, 0, stream>>>(
        hin, src, dst, ew, agg, di, csh);
    dim3 grid(NN / 128, dO / 64);
    gemm_wmma_kernel<<<grid, 256, 0, stream>>>(agg, hin, wr, wroot, bias,
                                               hout, di, dO);
  };
  auto bn = [&](float* h, int dO, const float* g, const float* b) {
    colstats_kernel<<<dO, 256, 0, stream>>>(h, mean, var, dO);
    long long tot = (long long)NN * dO;
    bnrelu_kernel<<<(int)((tot + 255) / 256), 256, 0, stream>>>(h, g, b,
                                                                mean, var, dO);
  };

  // encoder
  gconv(x,  hA, 256, 256, enc0wr, enc0ws, enc0b); bn(hA, 256, ebn0g, ebn0b);
  gconv(hA, hB, 256, 128, enc1wr, enc1ws, enc1b); bn(hB, 128, ebn1g, ebn1b);
  gconv(hB, hA, 128, 64,  enc2wr, enc2ws, enc2b);
  // conv codec
  conv_pipe_kernel<<<NN, 64, 0, stream>>>(hA, hB, we0, be0, we1, be1, we2, be2,
                                          wd0, bd0, wd1, bd1, wd2, bd2);
  // decoder
  gconv(hB, hA, 64,  128, dec0wr, dec0ws, dec0b); bn(hA, 128, dbn0g, dbn0b);
  gconv(hA, hB, 128, 256, dec1wr, dec1ws, dec1b); bn(hB, 256, dbn1g, dbn1b);
  gconv(hB, ho, 256, 256, dec2wr, dec2ws, dec2b);
}